// Renderer_47845935678015
// MI455X (gfx1250) — compile-verified
//
#include <hip/hip_runtime.h>
#include <stdint.h>

// ---------------------------------------------------------------------------
// Barycentric attribute interpolation (renderer) for MI455X / gfx1250.
//
// Bandwidth-bound kernel (~380MB HBM traffic -> ~16us roofline @ 23.3TB/s).
// CDNA5 paths used: async global->LDS staging (ASYNCcnt), s_wait_asynccnt,
// non-temporal (TH=NT) streaming stores, wave32-shaped 4-lanes-per-pixel
// layout for fully coalesced b128 stores.
// ---------------------------------------------------------------------------

typedef __attribute__((ext_vector_type(4))) float v4f;
typedef __attribute__((ext_vector_type(4))) int   v4i;

// Builtin signature (from hipcc diagnostic): first param is
// "int __vector_size__(4) __device__ *"  ==  int4 in address_space(1).
typedef __attribute__((address_space(1))) v4i* g_v4i_ptr;  // global int4*
typedef __attribute__((address_space(3))) v4i* l_v4i_ptr;  // LDS    int4*

#define B_    2
#define H_    1080
#define W_    1920
#define V_    100000
#define F_    200000
#define D_    16
#define HW_   (H_ * W_)
#define NPIX  (B_ * HW_)          // 4,147,200
#define PIX_PER_BLOCK 64          // 256 threads, 4 lanes per pixel
#define NBLOCKS (NPIX / PIX_PER_BLOCK)  // 64,800 exactly (no tail)

__device__ __forceinline__ void async_copy_b128(const void* g, void* l) {
#if defined(__has_builtin) && __has_builtin(__builtin_amdgcn_global_load_async_to_lds_b128)
  __builtin_amdgcn_global_load_async_to_lds_b128(
      (g_v4i_ptr)(uintptr_t)g,
      (l_v4i_ptr)(uint32_t)(uintptr_t)l,
      /*offset=*/0, /*cpol=*/0);
#else
  asm volatile("global_load_async_to_lds_b128 %0, %1, off"
               :
               : "v"((uint32_t)(uintptr_t)(l)), "v"((uint64_t)(uintptr_t)(g))
               : "memory");
#endif
}

__device__ __forceinline__ void wait_asynccnt0() {
#if defined(__has_builtin) && __has_builtin(__builtin_amdgcn_s_wait_asynccnt)
  __builtin_amdgcn_s_wait_asynccnt(0);
#else
  asm volatile("s_wait_asynccnt 0x0" ::: "memory");
#endif
}

__global__ __launch_bounds__(256) void renderer_interp_kernel(
    const float* __restrict__ attributes,  // (B, V, D)
    const float* __restrict__ bary,        // (B, H, W, 3)
    const int*   __restrict__ faces,       // (F, 3)
    const int*   __restrict__ ptf,         // (B, H, W)
    float* __restrict__ out,               // (B, H, W, D)
    float* __restrict__ mask)              // (B, H, W) as 1.0f / 0.0f
{
  __shared__ int   s_ptf[PIX_PER_BLOCK];           // 256 B
  __shared__ float s_bary[PIX_PER_BLOCK * 3];      // 768 B

  const int tid    = threadIdx.x;
  const int block0 = blockIdx.x * PIX_PER_BLOCK;   // first pixel of block

  // ---- Stage this block's pix_to_face (256B) and bary (768B) into LDS ----
  // Lanes 0..15 copy ptf, lanes 16..63 copy bary; 16B per lane, ASYNCcnt.
  if (tid < 16) {
    const char* g = (const char*)(ptf + block0) + tid * 16;
    async_copy_b128(g, (char*)s_ptf + tid * 16);
  } else if (tid < 64) {
    const char* g = (const char*)(bary + (size_t)block0 * 3) + (tid - 16) * 16;
    async_copy_b128(g, (char*)s_bary + (tid - 16) * 16);
  }
  wait_asynccnt0();       // issuing waves: data has landed in LDS
  __syncthreads();        // make LDS visible to all 8 waves

  // ---- Interpolate: 4 lanes per pixel, each lane owns a float4 of D=16 ----
  const int qid = tid & 3;        // which quarter of the 16 attributes
  const int pl  = tid >> 2;       // pixel-in-block 0..63
  const int p   = block0 + pl;    // global pixel id
  const int b   = (p >= HW_) ? 1 : 0;   // batch index (B=2)

  int f = s_ptf[pl];
  if (f < 0) f += F_;             // JAX/torch negative indexing: -1 -> F-1

  const int fi = 3 * f;
  const int v0 = faces[fi + 0];
  const int v1 = faces[fi + 1];
  const int v2 = faces[fi + 2];

  const float w0 = s_bary[3 * pl + 0];
  const float w1 = s_bary[3 * pl + 1];
  const float w2 = s_bary[3 * pl + 2];

  const float* abase = attributes + (size_t)b * ((size_t)V_ * D_) + qid * 4;
  const v4f a0 = *(const v4f*)(abase + (size_t)v0 * D_);
  const v4f a1 = *(const v4f*)(abase + (size_t)v1 * D_);
  const v4f a2 = *(const v4f*)(abase + (size_t)v2 * D_);

  v4f r = w0 * a0 + w1 * a1 + w2 * a2;

  // Streaming (non-temporal) b128 store: a wave writes 512 contiguous bytes.
  __builtin_nontemporal_store(r, (v4f*)(out + (size_t)p * D_ + qid * 4));

  // ---- Mask: lanes 0..63 write 256 contiguous bytes from staged ptf ----
  if (tid < PIX_PER_BLOCK) {
    const float m = (s_ptf[tid] != -1) ? 1.0f : 0.0f;
    __builtin_nontemporal_store(m, mask + block0 + tid);
  }
}

extern "C" void kernel_launch(void* const* d_in, const int* in_sizes, int n_in,
                              void* d_out, int out_size, void* d_ws, size_t ws_size,
                              hipStream_t stream) {
  (void)in_sizes; (void)n_in; (void)d_ws; (void)ws_size; (void)out_size;

  const float* attributes = (const float*)d_in[0];  // (B,V,D) f32
  const float* bary       = (const float*)d_in[1];  // (B,H,W,3) f32
  const int*   faces      = (const int*)  d_in[2];  // (F,3) i32
  const int*   ptf        = (const int*)  d_in[3];  // (B,H,W) i32

  float* out  = (float*)d_out;                      // (B,H,W,D)
  float* mask = out + (size_t)NPIX * D_;            // (B,H,W) appended

  renderer_interp_kernel<<<dim3(NBLOCKS), dim3(256), 0, stream>>>(
      attributes, bary, faces, ptf, out, mask);
}